// SAGEConv_43671227466484
// MI455X (gfx1250) — compile-verified
//
#include <hip/hip_runtime.h>

typedef __attribute__((ext_vector_type(2))) float v2f;
typedef __attribute__((ext_vector_type(8))) float v8f;

#define DIM 64

// ---------------------------------------------------------------------------
// Kernel 0: zero the workspace (agg[N*64] followed by deg[N])
// ---------------------------------------------------------------------------
__global__ void sage_zero_kernel(float* __restrict__ ws, int n) {
    int i = blockIdx.x * blockDim.x + threadIdx.x;
    if (i < n) ws[i] = 0.0f;
}

// ---------------------------------------------------------------------------
// Kernel 1: edge scatter.  One wave32 per edge: lanes cooperatively read the
// 64-float source row (contiguous 256B -> coalesced) and atomically add into
// the destination row.  x (12.8MB) and agg (12.8MB) are L2-resident on MI455X
// (192MB L2), so this runs at L2/atomic rate, not HBM.
// unsafeAtomicAdd lowers to non-returning global_atomic_add_f32 (STOREcnt).
// ---------------------------------------------------------------------------
__global__ void __launch_bounds__(256)
sage_edge_scatter_kernel(const float* __restrict__ x,
                         const int*   __restrict__ src,
                         const int*   __restrict__ dst,
                         float*       __restrict__ agg,
                         float*       __restrict__ deg,
                         int E) {
    int gid  = blockIdx.x * blockDim.x + threadIdx.x;
    int e    = gid >> 5;      // wave32 per edge
    int lane = gid & 31;
    if (e >= E) return;
    int s = src[e];
    int d = dst[e];
    const float2* xr = (const float2*)(x + (size_t)s * DIM);
    float2 v = xr[lane];                       // lane covers features 2*lane, 2*lane+1
    float* ar = agg + (size_t)d * DIM + lane * 2;
    unsafeAtomicAdd(ar,     v.x);
    unsafeAtomicAdd(ar + 1, v.y);
    if (lane == 0) unsafeAtomicAdd(deg + d, 1.0f);
}

// ---------------------------------------------------------------------------
// Kernel 2: fused  out = x@W_self + (agg/max(deg,1))@W_neigh + (b_self+b_neigh)
// via V_WMMA_F32_16X16X4_F32.
//
// Per wave: one 16-row x 64-col output tile, 4 v8f accumulators (one per
// 16-col N tile).  K=64 walked in steps of 4 for both GEMMs -> 128 WMMA ops
// per tile, all accumulating in fp32.
//
// f32 16x16x4 A layout (ISA 7.12.2): lane L: M = L&15, hi = L>>4,
//   a.x = A[M][kb+2*hi], a.y = A[M][kb+2*hi+1]
// B (4x16) mirrors it: b.x = B[kb+2*hi][N], b.y = B[kb+2*hi+1][N], N = L&15.
// C/D: acc[r] = D[r + 8*hi][L&15].
// ---------------------------------------------------------------------------
__global__ void __launch_bounds__(256)
sage_wmma_gemm_kernel(const float* __restrict__ x,
                      const float* __restrict__ agg,
                      const float* __restrict__ deg,
                      const float* __restrict__ Wself,
                      const float* __restrict__ bself,
                      const float* __restrict__ Wneigh,
                      const float* __restrict__ bneigh,
                      float*       __restrict__ out,
                      int nTiles) {
    __shared__ float sWs[DIM * DIM];   // W_self  64x64 row-major [k][o]
    __shared__ float sWn[DIM * DIM];   // W_neigh 64x64 row-major [k][o]
    __shared__ float sB [DIM];         // b_self + b_neigh

    // cooperative stage of the weight matrices + fused bias into LDS
    for (int i = threadIdx.x; i < DIM * DIM; i += 256) {
        sWs[i] = Wself[i];
        sWn[i] = Wneigh[i];
    }
    if (threadIdx.x < DIM) sB[threadIdx.x] = bself[threadIdx.x] + bneigh[threadIdx.x];
    __syncthreads();

    const int wave = threadIdx.x >> 5;
    const int lane = threadIdx.x & 31;
    const int tile = blockIdx.x * 8 + wave;
    if (tile >= nTiles) return;        // whole wave exits together: EXEC all-1s otherwise

    const int m    = lane & 15;        // A/C row-within-tile selector (and C column)
    const int hi   = lane >> 4;        // K-half selector
    const int row0 = tile * 16;
    const int row  = row0 + m;

    const float dinv = 1.0f / fmaxf(deg[row], 1.0f);
    const float* __restrict__ xrow = x   + (size_t)row * DIM;
    const float* __restrict__ arow = agg + (size_t)row * DIM;

    // init accumulators with the fused bias (depends only on output column)
    v8f acc[4];
    for (int nt = 0; nt < 4; ++nt) {
        float b = sB[nt * 16 + m];
        for (int r = 0; r < 8; ++r) acc[nt][r] = b;
    }

    // K loop over both GEMMs, accumulating into the same C
    for (int kb = 0; kb < DIM; kb += 4) {
        const int kk = kb + 2 * hi;
        v2f as, an;
        as.x = xrow[kk];
        as.y = xrow[kk + 1];
        an.x = arow[kk]     * dinv;    // h_neigh fused here
        an.y = arow[kk + 1] * dinv;

        for (int nt = 0; nt < 4; ++nt) {
            const int bi = kk * DIM + nt * 16 + m;
            v2f bs, bn;
            bs.x = sWs[bi];
            bs.y = sWs[bi + DIM];
            bn.x = sWn[bi];
            bn.y = sWn[bi + DIM];
            acc[nt] = __builtin_amdgcn_wmma_f32_16x16x4_f32(
                          false, as, false, bs, (short)0, acc[nt], false, false);
            acc[nt] = __builtin_amdgcn_wmma_f32_16x16x4_f32(
                          false, an, false, bn, (short)0, acc[nt], false, false);
        }
    }

    // store: lanes 0-15 cover cols 0-15 of row r, lanes 16-31 row r+8
    for (int nt = 0; nt < 4; ++nt) {
        const int col = nt * 16 + m;
        for (int r = 0; r < 8; ++r) {
            const int mrow = r + 8 * hi;
            out[(size_t)(row0 + mrow) * DIM + col] = acc[nt][r];
        }
    }
}

// ---------------------------------------------------------------------------
// launch
// ---------------------------------------------------------------------------
extern "C" void kernel_launch(void* const* d_in, const int* in_sizes, int n_in,
                              void* d_out, int out_size, void* d_ws, size_t ws_size,
                              hipStream_t stream) {
    const float* x      = (const float*)d_in[0];
    const int*   src    = (const int*)  d_in[1];
    const int*   dst    = (const int*)  d_in[2];
    const float* Wself  = (const float*)d_in[3];
    const float* bself  = (const float*)d_in[4];
    const float* Wneigh = (const float*)d_in[5];
    const float* bneigh = (const float*)d_in[6];
    float*       out    = (float*)d_out;

    const int N = in_sizes[0] / DIM;   // 50000
    const int E = in_sizes[1];         // 800000

    float* agg = (float*)d_ws;                 // [N*64]
    float* deg = agg + (size_t)N * DIM;        // [N]

    // 1) zero agg + deg
    const int zn = N * DIM + N;
    sage_zero_kernel<<<(zn + 255) / 256, 256, 0, stream>>>(agg, zn);

    // 2) edge scatter (one wave32 per edge; 8 edges per 256-thread block)
    const long long totThreads = (long long)E * 32;
    sage_edge_scatter_kernel<<<(int)((totThreads + 255) / 256), 256, 0, stream>>>(
        x, src, dst, agg, deg, E);

    // 3) fused normalize + dual GEMM + bias via WMMA
    const int nTiles = (N + 15) / 16;          // 3125, exact
    sage_wmma_gemm_kernel<<<(nTiles + 7) / 8, 256, 0, stream>>>(
        x, agg, deg, Wself, bself, Wneigh, bneigh, out, nTiles);
}